// ODEINDLayer_84920093377148
// MI455X (gfx1250) — compile-verified
//
#include <hip/hip_runtime.h>
#include <hip/hip_bf16.h>
#include <math.h>

// MI455X / gfx1250, wave32. One workgroup (1 wave) per batch problem.
// Block-tridiagonal normal equations; D-block Gram via V_WMMA_F32_16X16X4_F32,
// factorization/solve in f64 entirely in LDS.

typedef __attribute__((ext_vector_type(2))) float v2f;
typedef __attribute__((ext_vector_type(8))) float v8f;

#define NSTEP 100
#define T1    99
#define NB    256          // BS * N_IND = 32 * 8
#define OUT_U   (NB * NSTEP)          // 25600 per component
#define OUT_EPS (3 * OUT_U)           // 76800
#define OUT_ST  (OUT_EPS + NB)        // 77056

// E = [Dt | Dt1] : 5 x 6.  Every entry is a + b*h + c*h^2 with constant (a,b,c):
//   row0: [ 1,  h, h^2/2 | -1,  0,  0    ]
//   row1: [ 0,  1,  h    |  0, -1,  0    ]
//   row2: [-1,  0,  0    |  1, -h, h^2/2 ]
//   row3: [ 0, -1,  0    |  0,  1, -h    ]
//   row4: [ 0, -1, -h/2  |  0,  1, -h/2  ]
// Returns the (a,b,c) polynomial coefficients for E[r][c]; r<0 or c>=6 -> zeros.
__device__ __forceinline__ void Ecoef(int r, int c, float* pa, float* pb, float* pc) {
  float a = 0.f, b = 0.f, q = 0.f;
  if (r == 0)      { if (c == 0) a = 1.f;  else if (c == 1) b = 1.f;   else if (c == 2) q = 0.5f; else if (c == 3) a = -1.f; }
  else if (r == 1) { if (c == 1) a = 1.f;  else if (c == 2) b = 1.f;   else if (c == 4) a = -1.f; }
  else if (r == 2) { if (c == 0) a = -1.f; else if (c == 3) a = 1.f;   else if (c == 4) b = -1.f; else if (c == 5) q = 0.5f; }
  else if (r == 3) { if (c == 1) a = -1.f; else if (c == 4) a = 1.f;   else if (c == 5) b = -1.f; }
  else if (r == 4) { if (c == 1) a = -1.f; else if (c == 2) b = -0.5f; else if (c == 4) a = 1.f;  else if (c == 5) b = -0.5f; }
  *pa = a; *pb = b; *pc = q;
}

// 3x3 SPD Cholesky in packed-lower storage {00,10,11,20,21,22}, in place.
__device__ __forceinline__ void chol3(double* s) {
  double l00 = sqrt(s[0]);
  double l10 = s[1] / l00;
  double l11 = sqrt(s[2] - l10 * l10);
  double l20 = s[3] / l00;
  double l21 = (s[4] - l20 * l10) / l11;
  double l22 = sqrt(s[5] - l20 * l20 - l21 * l21);
  s[0] = l00; s[1] = l10; s[2] = l11; s[3] = l20; s[4] = l21; s[5] = l22;
}
__device__ __forceinline__ void lsolve3(const double* l, double* v) {  // L y = v
  double v0 = v[0] / l[0];
  double v1 = (v[1] - l[1] * v0) / l[2];
  double v2 = (v[2] - l[3] * v0 - l[4] * v1) / l[5];
  v[0] = v0; v[1] = v1; v[2] = v2;
}
__device__ __forceinline__ void ltsolve3(const double* l, double* v) { // L^T x = v
  double x2 = v[2] / l[5];
  double x1 = (v[1] - l[4] * x2) / l[2];
  double x0 = (v[0] - l[1] * x1 - l[3] * x2) / l[0];
  v[0] = x0; v[1] = x1; v[2] = x2;
}

__global__ __launch_bounds__(32) void ode_blocktri_kernel(
    const float* __restrict__ coeffs,   // (256, 100, 3)
    const float* __restrict__ rhs,      // (256, 100)
    const float* __restrict__ ivr,      // (256, 2)
    const float* __restrict__ steps,    // (256, 99)
    float* __restrict__ out)            // u0|u1|u2|eps|steps concat
{
  __shared__ double S[NSTEP][6];   // diag blocks (packed lower), later L_t
  __shared__ double Bo[T1][9];     // off-diag blocks B_t (row-major), later M_t
  __shared__ double gv[NSTEP][3];  // Atb, later y, later x
  __shared__ float  hstep[T1];

  const int b = blockIdx.x;
  const int lane = threadIdx.x;

  // ---- Phase 0: zero LDS, stage step sizes ----
  for (int i = lane; i < NSTEP * 6; i += 32) ((double*)S)[i] = 0.0;
  for (int i = lane; i < T1 * 9;   i += 32) ((double*)Bo)[i] = 0.0;
  for (int i = lane; i < NSTEP * 3; i += 32) ((double*)gv)[i] = 0.0;
  for (int i = lane; i < T1; i += 32) hstep[i] = steps[b * T1 + i];
  __syncthreads();

  // ---- Phase 1: Eq rows (c c^T) + initial-value rows, f64 ----
  for (int t = lane; t < NSTEP; t += 32) {
    const double c0 = (double)coeffs[(b * NSTEP + t) * 3 + 0];
    const double c1 = (double)coeffs[(b * NSTEP + t) * 3 + 1];
    const double c2 = (double)coeffs[(b * NSTEP + t) * 3 + 2];
    const double r  = (double)rhs[b * NSTEP + t];
    S[t][0] += c0 * c0; S[t][1] += c1 * c0; S[t][2] += c1 * c1;
    S[t][3] += c2 * c0; S[t][4] += c2 * c1; S[t][5] += c2 * c2;
    gv[t][0] += c0 * r; gv[t][1] += c1 * r; gv[t][2] += c2 * r;
    if (t == 0) {                      // Iv rows: unit rows on (0,0) and (0,1)
      S[0][0] += 1.0; S[0][2] += 1.0;
      gv[0][0] += (double)ivr[b * 2 + 0];
      gv[0][1] += (double)ivr[b * 2 + 1];
    }
  }
  __syncthreads();

  // ---- Phase 2: D-block Gram G = E^T E (6x6) via WMMA f32 16x16x4 ----
  // A (16x4): lanes 0-15 hold K0/K1 in comps 0/1; lanes 16-31 hold K2/K3. M = lane&15.
  // B (4x16): same K striping, N = lane&15.  A and B fragments are identical here.
  // C/D (16x16): G[i][j] lands in lane j, component i (i < 8).
  //
  // Each lane's (k, m) is fixed, so hoist the E-entry polynomial coefficients
  // (a + b*h + c*h^2) out of the loop: the loop body is 6 FMAs + 2 WMMAs + DS stores.
  const int m = lane & 15;
  const int khalf = lane >> 4;
  const int col = (m < 6) ? m : 6;          // out-of-tile columns -> zero coefs
  float a0, b0, q0, a1, b1, q1, a2, b2, q2;
  Ecoef(2 * khalf,                 col, &a0, &b0, &q0);   // chunk0 comp x: row 2*khalf
  Ecoef(2 * khalf + 1,             col, &a1, &b1, &q1);   // chunk0 comp y: row 2*khalf+1
  Ecoef((khalf == 0) ? 4 : -1,     col, &a2, &b2, &q2);   // chunk1 comp x: row 4 (else 0)

  for (int p = 0; p < T1; ++p) {
    const float h = hstep[p];
    v2f f0, f1;
    f0.x = fmaf(fmaf(q0, h, b0), h, a0);
    f0.y = fmaf(fmaf(q1, h, b1), h, a1);
    f1.x = fmaf(fmaf(q2, h, b2), h, a2);
    f1.y = 0.f;
    v8f acc = {};
    acc = __builtin_amdgcn_wmma_f32_16x16x4_f32(false, f0, false, f0,
                                                (short)0, acc, false, false);
    acc = __builtin_amdgcn_wmma_f32_16x16x4_f32(false, f1, false, f1,
                                                (short)0, acc, false, false);
    if (lane < 3) {                       // G00 = Dt^T Dt  -> S[p] (lower)
      const int j = lane;
      #pragma unroll
      for (int i = 0; i < 3; ++i)
        if (i >= j) S[p][i * (i + 1) / 2 + j] += (double)acc[i];
    } else if (lane < 6) {                // G01 -> B_p,  G11 -> S[p+1]
      const int j = lane - 3;
      #pragma unroll
      for (int i = 0; i < 3; ++i) Bo[p][i * 3 + j] += (double)acc[i];
      #pragma unroll
      for (int i = 0; i < 3; ++i)
        if (i >= j) S[p + 1][i * (i + 1) / 2 + j] += (double)acc[3 + i];
    }
  }
  __syncthreads();

  // ---- Phase 3: sequential block-tridiagonal Cholesky + solve (f64) ----
  if (lane == 0) {
    chol3(S[0]);
    lsolve3(S[0], gv[0]);
    for (int p = 0; p < T1; ++p) {
      const double* L = S[p];
      double X[3][3];                    // M_p: X L_p^T = B_p^T
      #pragma unroll
      for (int r = 0; r < 3; ++r) {
        const double c0 = Bo[p][0 + r];      // (B^T)[r][0] = B[0][r]
        const double c1 = Bo[p][3 + r];
        const double c2 = Bo[p][6 + r];
        const double x0 = c0 / L[0];
        const double x1 = (c1 - x0 * L[1]) / L[2];
        const double x2 = (c2 - x0 * L[3] - x1 * L[4]) / L[5];
        X[r][0] = x0; X[r][1] = x1; X[r][2] = x2;
      }
      #pragma unroll
      for (int r = 0; r < 3; ++r) {
        Bo[p][r * 3 + 0] = X[r][0]; Bo[p][r * 3 + 1] = X[r][1]; Bo[p][r * 3 + 2] = X[r][2];
      }
      // S[p+1] -= X X^T (lower), then factor
      S[p + 1][0] -= X[0][0]*X[0][0] + X[0][1]*X[0][1] + X[0][2]*X[0][2];
      S[p + 1][1] -= X[1][0]*X[0][0] + X[1][1]*X[0][1] + X[1][2]*X[0][2];
      S[p + 1][2] -= X[1][0]*X[1][0] + X[1][1]*X[1][1] + X[1][2]*X[1][2];
      S[p + 1][3] -= X[2][0]*X[0][0] + X[2][1]*X[0][1] + X[2][2]*X[0][2];
      S[p + 1][4] -= X[2][0]*X[1][0] + X[2][1]*X[1][1] + X[2][2]*X[1][2];
      S[p + 1][5] -= X[2][0]*X[2][0] + X[2][1]*X[2][1] + X[2][2]*X[2][2];
      chol3(S[p + 1]);
      // forward: y_{p+1} = L_{p+1}^{-1} (g_{p+1} - M_p y_p)
      const double y0 = gv[p][0], y1 = gv[p][1], y2 = gv[p][2];
      gv[p + 1][0] -= X[0][0]*y0 + X[0][1]*y1 + X[0][2]*y2;
      gv[p + 1][1] -= X[1][0]*y0 + X[1][1]*y1 + X[1][2]*y2;
      gv[p + 1][2] -= X[2][0]*y0 + X[2][1]*y1 + X[2][2]*y2;
      lsolve3(S[p + 1], gv[p + 1]);
    }
    // backward: x_p = L_p^{-T} (y_p - M_p^T x_{p+1})
    ltsolve3(S[T1], gv[T1]);
    for (int p = T1 - 1; p >= 0; --p) {
      const double x0 = gv[p + 1][0], x1 = gv[p + 1][1], x2 = gv[p + 1][2];
      gv[p][0] -= Bo[p][0]*x0 + Bo[p][3]*x1 + Bo[p][6]*x2;
      gv[p][1] -= Bo[p][1]*x0 + Bo[p][4]*x1 + Bo[p][7]*x2;
      gv[p][2] -= Bo[p][2]*x0 + Bo[p][5]*x1 + Bo[p][8]*x2;
      ltsolve3(S[p], gv[p]);
    }
  }
  __syncthreads();

  // ---- Phase 4: outputs (u0|u1|u2|eps|steps), all lanes ----
  for (int t = lane; t < NSTEP; t += 32) {
    out[0 * OUT_U + b * NSTEP + t] = (float)gv[t][0];
    out[1 * OUT_U + b * NSTEP + t] = (float)gv[t][1];
    out[2 * OUT_U + b * NSTEP + t] = (float)gv[t][2];
  }
  if (lane == 0) out[OUT_EPS + b] = 0.0f;
  for (int p = lane; p < T1; p += 32) out[OUT_ST + b * T1 + p] = hstep[p];
}

extern "C" void kernel_launch(void* const* d_in, const int* in_sizes, int n_in,
                              void* d_out, int out_size, void* d_ws, size_t ws_size,
                              hipStream_t stream) {
  (void)in_sizes; (void)n_in; (void)out_size; (void)d_ws; (void)ws_size;
  const float* coeffs = (const float*)d_in[0];
  const float* rhs    = (const float*)d_in[1];
  const float* ivr    = (const float*)d_in[2];
  const float* steps  = (const float*)d_in[3];
  float* out = (float*)d_out;
  hipLaunchKernelGGL(ode_blocktri_kernel, dim3(NB), dim3(32), 0, stream,
                     coeffs, rhs, ivr, steps, out);
}